// buhadanoAttention_60395830116510
// MI455X (gfx1250) — compile-verified
//
#include <hip/hip_runtime.h>
#include <hip/hip_bf16.h>

#define B_ 64
#define T_ 2048
#define D_ 512
#define U_ 10

typedef float v2f __attribute__((ext_vector_type(2)));
typedef float v8f __attribute__((ext_vector_type(8)));

// -------- K1: comb[b][u] = H[b] @ W_h[:,u] + b_h[u] + b_eo[u] (u<10), zero pad to 16
__global__ void k_comb(const float* __restrict__ H, const float* __restrict__ Wh,
                       const float* __restrict__ bh, const float* __restrict__ beo,
                       float* __restrict__ comb) {
  int b = blockIdx.x;
  int u = threadIdx.x;            // 32 lanes
  if (u >= 16) return;
  float acc = 0.f;
  if (u < U_) {
    acc = bh[u] + beo[u];
    const float* h = H + b * D_;
    for (int d = 0; d < D_; ++d)
      acc += h[d] * Wh[d * U_ + u];
  }
  comb[b * 16 + u] = acc;
}

// -------- K2: scores[b][t] = tanh(Eo@W_eo + comb) . W_s  via V_WMMA_F32_16X16X4_F32
__global__ void __launch_bounds__(256)
k_scores(const float* __restrict__ Eo, const float* __restrict__ Weo,
         const float* __restrict__ comb, const float* __restrict__ Ws,
         float* __restrict__ scores) {
  __shared__ float Wraw[D_ * U_];         // 20 KB raw W_eo, filled by async-to-LDS
  __shared__ float Wlds[D_ * 16];         // 32 KB, [d][16], N-padded with zeros
  const int tid = threadIdx.x;
  const int b = blockIdx.y;

  // Stage raw W_eo via GLOBAL_LOAD_ASYNC_TO_LDS_B128 (ASYNCcnt): 5 passes of
  // 256 lanes x 16B = 20480 B, fully coalesced, no VGPR return data.
  {
    unsigned lbase = (unsigned)(unsigned long long)(&Wraw[0]); // low 32b = LDS offset
    unsigned long long gbase = (unsigned long long)Weo;
    #pragma unroll
    for (int p = 0; p < 5; ++p) {
      unsigned byteoff = (unsigned)(p * 256 + tid) * 16u;
      unsigned ldst = lbase + byteoff;
      unsigned long long ga = gbase + byteoff;
      asm volatile("global_load_async_to_lds_b128 %0, %1, off"
                   :: "v"(ldst), "v"(ga) : "memory");
    }
    asm volatile("s_wait_asynccnt 0x0" ::: "memory");
  }
  __syncthreads();

  // Expand to zero-padded [512][16] layout (LDS -> LDS).
  for (int i = tid; i < D_ * 16; i += 256) {
    int u = i & 15;
    Wlds[i] = (u < U_) ? Wraw[(i >> 4) * U_ + u] : 0.f;
  }
  __syncthreads();

  const int wave = tid >> 5, lane = tid & 31;
  const int half = lane >> 4, l16 = lane & 15;
  const int t0 = (blockIdx.x * 8 + wave) * 16;
  const float* Arow = Eo + ((size_t)b * T_ + (size_t)(t0 + l16)) * D_;

  v8f c = {};
  // A 16x4 f32 layout: lanes 0-15 hold K=0(v0),K=1(v1); lanes 16-31 hold K=2,K=3 (same rows).
  // B 4x16 analogous: lanes hold column N=l16, same K split.
  #pragma unroll 8
  for (int k0 = 0; k0 < D_ / 4; ++k0) {
    const int kb = k0 * 4 + half * 2;
    v2f a = *(const v2f*)(Arow + kb);                 // global_load_b64, 8B aligned
    v2f w;
    w.x = Wlds[kb * 16 + l16];                        // conflict-free ds_load_b32
    w.y = Wlds[(kb + 1) * 16 + l16];
    c = __builtin_amdgcn_wmma_f32_16x16x4_f32(false, a, false, w,
                                              (short)0, c, false, false);
  }

  const float combv = comb[b * 16 + l16];
  const float wsv = (l16 < U_) ? Ws[l16] : 0.f;

  // C layout: lane<16 holds rows 0..7 (VGPR j = row j) of column l16; lane>=16 rows 8..15.
  float sv[8];
  #pragma unroll
  for (int j = 0; j < 8; ++j) {
    float v = tanhf(c[j] + combv) * wsv;
    #pragma unroll
    for (int m = 1; m < 16; m <<= 1) v += __shfl_xor(v, m, 32);   // sum over 16 columns
    sv[j] = v;
  }
  if (l16 == 0) {
    float* so = scores + (size_t)b * T_ + t0 + half * 8;
    #pragma unroll
    for (int j = 0; j < 8; ++j) so[j] = sv[j];
  }
}

// -------- K3: softmax over T per batch; writes weight output
__global__ void __launch_bounds__(256)
k_softmax(const float* __restrict__ scores, float* __restrict__ weight) {
  __shared__ float red[256];
  const int b = blockIdx.x, tid = threadIdx.x;
  const float* s = scores + (size_t)b * T_;
  float loc[8];
  float mx = -3.402823466e38f;
  #pragma unroll
  for (int i = 0; i < 8; ++i) { loc[i] = s[tid + i * 256]; mx = fmaxf(mx, loc[i]); }
  red[tid] = mx; __syncthreads();
  for (int off = 128; off > 0; off >>= 1) {
    if (tid < off) red[tid] = fmaxf(red[tid], red[tid + off]);
    __syncthreads();
  }
  mx = red[0]; __syncthreads();
  float sum = 0.f;
  #pragma unroll
  for (int i = 0; i < 8; ++i) { loc[i] = __expf(loc[i] - mx); sum += loc[i]; }
  red[tid] = sum; __syncthreads();
  for (int off = 128; off > 0; off >>= 1) {
    if (tid < off) red[tid] += red[tid + off];
    __syncthreads();
  }
  const float inv = 1.f / red[0];
  float* w = weight + (size_t)b * T_;
  #pragma unroll
  for (int i = 0; i < 8; ++i) w[tid + i * 256] = loc[i] * inv;
}

// -------- K4: context[b][d] = sum_t Eo[b][t][d] * weight[b][t]
__global__ void __launch_bounds__(64)
k_context(const float* __restrict__ Eo, const float* __restrict__ weight,
          float* __restrict__ ctx) {
  const int b = blockIdx.y;
  const int d = blockIdx.x * 64 + threadIdx.x;
  const float* e = Eo + (size_t)b * T_ * D_ + d;
  const float* w = weight + (size_t)b * T_;
  float a0 = 0.f, a1 = 0.f, a2 = 0.f, a3 = 0.f;
  for (int t = 0; t < T_; t += 4) {
    // Pull the cacheline 32 rows (64 KB) ahead; speculative prefetch past the
    // buffer end is silently dropped (ISA 10.5). Emits global_prefetch_b8.
    __builtin_prefetch(e + (size_t)(t + 32) * D_, 0, 0);
    a0 += e[(size_t)(t + 0) * D_] * w[t + 0];
    a1 += e[(size_t)(t + 1) * D_] * w[t + 1];
    a2 += e[(size_t)(t + 2) * D_] * w[t + 2];
    a3 += e[(size_t)(t + 3) * D_] * w[t + 3];
  }
  ctx[b * D_ + d] = (a0 + a1) + (a2 + a3);
}

extern "C" void kernel_launch(void* const* d_in, const int* in_sizes, int n_in,
                              void* d_out, int out_size, void* d_ws, size_t ws_size,
                              hipStream_t stream) {
  const float* Eo  = (const float*)d_in[0];
  const float* H   = (const float*)d_in[1];
  const float* Weo = (const float*)d_in[2];
  const float* beo = (const float*)d_in[3];
  const float* Wh  = (const float*)d_in[4];
  const float* bh  = (const float*)d_in[5];
  const float* Ws  = (const float*)d_in[6];
  // d_in[7] = b_s: uniform shift of all scores -> softmax-invariant -> omitted.

  float* ctx    = (float*)d_out;            // [B, D]
  float* weight = ctx + B_ * D_;            // [B, T, 1]

  float* comb   = (float*)d_ws;             // [B, 16]
  float* scores = comb + B_ * 16;           // [B, T]

  k_comb<<<B_, 32, 0, stream>>>(H, Wh, bh, beo, comb);
  k_scores<<<dim3(T_ / 128, B_), 256, 0, stream>>>(Eo, Weo, comb, Ws, scores);
  k_softmax<<<B_, 256, 0, stream>>>(scores, weight);
  k_context<<<dim3(D_ / 64, B_), 64, 0, stream>>>(Eo, weight, ctx);
}